// E2EAverage_53455162966648
// MI455X (gfx1250) — compile-verified
//
#include <hip/hip_runtime.h>

// MI455X / gfx1250 (CDNA5, wave32). Batched dot-product (GEMV per batch)
// mi[b,k] = dot(samples[b,k,:], anchors[b,:]) / T
// HBM-bound: 537 MB streamed once -> ~23 us floor @ 23.3 TB/s.
// fp32-exact via V_WMMA_F32_16X16X4_F32; no predicated loads (row clamp),
// all 32 A-fragment loads issued before the WMMA chain for deep MLP.

typedef float v2f __attribute__((ext_vector_type(2)));
typedef float v8f __attribute__((ext_vector_type(8)));

#define NB 256                    // batch size
#define NK 4097                   // K+1 samples per batch
#define NF 128                    // feature dim
#define NCHUNK (NF / 4)           // 32 K-chunks of 4 per dot product
#define INV_T (1.0f / 0.07f)
#define WAVES_PER_BLOCK 8
#define BLOCKS_PER_BATCH 8
#define WAVES_PER_BATCH (WAVES_PER_BLOCK * BLOCKS_PER_BATCH)  // 64
#define NTILES ((NK + 15) / 16)   // 257 k-tiles of 16 rows per batch

__global__ __launch_bounds__(256)
void mi_dot_wmma_kernel(const float* __restrict__ feat, float* __restrict__ out) {
    __shared__ float sAnchor[NF];

    const int b   = blockIdx.y;
    const int tid = threadIdx.x;

    const float* anchors = feat;                        // [NB][NF]
    const float* samples = feat + (size_t)NB * NF;      // [NB][NK][NF]

    // Stage this batch's anchor row (512 B) into LDS once per block.
    if (tid < NF) sAnchor[tid] = anchors[(size_t)b * NF + tid];
    __syncthreads();

    const int wave = tid >> 5;
    const int lane = tid & 31;
    const int m    = lane & 15;            // A-matrix row (M) held by this lane
    const int koff = (lane >> 4) << 1;     // K sub-pair: 0 (lanes 0-15), 2 (lanes 16-31)

    // B fragments: 4x16 fp32 anchor chunks broadcast across all 16 N columns.
    // Value depends only on this lane's K rows -> anchor[4c+koff .. +1].
    // Loaded from LDS into registers ONCE per wave, reused for every tile.
    v2f bf[NCHUNK];
    #pragma unroll
    for (int c = 0; c < NCHUNK; ++c) {
        bf[c].x = sAnchor[4 * c + koff];
        bf[c].y = sAnchor[4 * c + koff + 1];
    }

    const float* sbase = samples + (size_t)b * NK * NF;

    // Grid-stride over this batch's 257 k-tiles (wave-uniform loop:
    // EXEC stays all-1s, as WMMA requires).
    for (int tile = blockIdx.x * WAVES_PER_BLOCK + wave; tile < NTILES;
         tile += WAVES_PER_BATCH) {
        const int k0  = tile * 16;
        int row = k0 + m;
        if (row > NK - 1) row = NK - 1;    // clamp tail rows in-bounds:
                                           // garbage rows computed, never stored
        const float* rowp = sbase + (size_t)row * NF + koff;

        // Issue ALL 32 A-fragment loads before any WMMA consumes them:
        // keeps up to 32 NT loads in flight per wave (MLP for 23.3 TB/s).
        v2f a[NCHUNK];
        #pragma unroll
        for (int c = 0; c < NCHUNK; ++c) {
            a[c] = __builtin_nontemporal_load((const v2f*)(rowp + 4 * c));
        }

        // K = 128 = 32 chained fp32 WMMAs accumulating in C/D.
        v8f acc = {0.f, 0.f, 0.f, 0.f, 0.f, 0.f, 0.f, 0.f};
        #pragma unroll
        for (int c = 0; c < NCHUNK; ++c) {
            acc = __builtin_amdgcn_wmma_f32_16x16x4_f32(
                /*neg_a=*/false, a[c], /*neg_b=*/false, bf[c],
                /*c_mod=*/(short)0, acc, /*reuse_a=*/false, /*reuse_b=*/false);
        }

        // D layout (16x16 f32): lane (N=lane%16), VGPR r -> M = r + 8*(lane>=16).
        // All N columns identical; column-0 lanes (0 and 16) write 8 rows each.
        if ((lane & 15) == 0) {
            const int mbase = (lane >> 4) * 8;
            #pragma unroll
            for (int r = 0; r < 8; ++r) {
                const int krow = k0 + mbase + r;
                if (krow < NK) {
                    __builtin_nontemporal_store(acc[r] * INV_T,
                                                &out[(size_t)b * NK + krow]);
                }
            }
        }
    }
}

extern "C" void kernel_launch(void* const* d_in, const int* in_sizes, int n_in,
                              void* d_out, int out_size, void* d_ws, size_t ws_size,
                              hipStream_t stream) {
    const float* feat = (const float*)d_in[0];
    float* out = (float*)d_out;

    dim3 grid(BLOCKS_PER_BATCH, NB, 1);   // (8, 256) blocks
    dim3 block(256, 1, 1);                // 8 waves; each wave strides over k-tiles
    mi_dot_wmma_kernel<<<grid, block, 0, stream>>>(feat, out);
}